// Mamba2MCSelectLMHeadModel_55972013801601
// MI455X (gfx1250) — compile-verified
//
#include <hip/hip_runtime.h>
#include <hip/hip_bf16.h>
#include <cstdint>

// ---------------- model constants (from the reference) ----------------
#define SEGSZ      64
#define MAXCACHED  16
#define TOPK       8
#define CAPN       17          // MAX_CACHED + 1
#define THRESH     (-1.0f)
#define NEGV       (-1e30f)

typedef float    v2f  __attribute__((ext_vector_type(2)));
typedef float    v8f  __attribute__((ext_vector_type(8)));
typedef uint32_t u32x4 __attribute__((ext_vector_type(4)));
typedef int      i32x4 __attribute__((ext_vector_type(4)));
typedef int      i32x8 __attribute__((ext_vector_type(8)));

#ifndef __has_builtin
#define __has_builtin(x) 0
#endif

#if __has_builtin(__builtin_amdgcn_tensor_load_to_lds) && \
    __has_builtin(__builtin_amdgcn_s_wait_tensorcnt)
#define HAVE_TDM 1
#else
#define HAVE_TDM 0
#endif

__device__ __forceinline__ v8f wmma_f32_16x16x4(v2f a, v2f b, v8f c) {
#if __has_builtin(__builtin_amdgcn_wmma_f32_16x16x4_f32)
  // (neg_a, A, neg_b, B, c_mod, C, reuse_a, reuse_b)
  return __builtin_amdgcn_wmma_f32_16x16x4_f32(false, a, false, b, (short)0, c,
                                               false, false);
#else
  c[0] += a[0] * b[0];
  return c;
#endif
}

#if HAVE_TDM
// ---------------------------------------------------------------------
// TDM: load a 2-D tile (tile_dim0 = 32 dwords along contiguous axis,
// tile_dim1 = 128 rows) from global into LDS, padding 4 dwords after
// every 32 (pad_interval code 4, pad_amount code 3) -> LDS row stride 36
// dwords, matching LDST below. Remaining-extent tensor dims give
// OOB-reads-as-zero for the ragged last lm_w tile.  D# packing per
// CDNA5 ISA.
// ---------------------------------------------------------------------
__device__ __forceinline__ void tdm_load_tile_2d(uint32_t lds_byte_off,
                                                 const float* tile_ptr,
                                                 uint32_t d0_rem, uint32_t d1_rem,
                                                 uint32_t stride0_elems) {
  const uint64_t ga = (uint64_t)(uintptr_t)tile_ptr;
  u32x4 g0;
  g0[0] = 1u;                                        // count=1, user descriptor
  g0[1] = lds_byte_off;                              // lds_addr
  g0[2] = (uint32_t)(ga & 0xFFFFFFFFu);              // global_addr[31:0]
  g0[3] = (uint32_t)((ga >> 32) & 0x1FFFFFFu)        // global_addr[56:32]
        | (2u << 30);                                // type = 2 ("image")
  i32x8 g1;
  g1[0] = (int)((2u << 16)                           // data_size = 4B
              | (1u << 20)                           // pad_enable
              | (4u << 22)                           // pad_interval: 32 dwords
              | (3u << 25));                         // pad_amount: 4 dwords
  g1[1] = (int)((d0_rem & 0xFFFFu) << 16);           // tensor_dim0[15:0]
  g1[2] = (int)(((d0_rem >> 16) & 0xFFFFu)           // tensor_dim0[31:16]
              | ((d1_rem & 0xFFFFu) << 16));         // tensor_dim1[15:0]
  g1[3] = (int)(((d1_rem >> 16) & 0xFFFFu)           // tensor_dim1[31:16]
              | (32u << 16));                        // tile_dim0 = 32
  g1[4] = (int)(128u);                               // tile_dim1 = 128, tile_dim2 = 0
  g1[5] = (int)stride0_elems;                        // tensor_dim0_stride[31:0]
  g1[6] = 0;                                         // stride0[47:32], stride1 lo
  g1[7] = 0;
  i32x4 g2 = {};
  i32x4 g3 = {};
#if __clang_major__ >= 23
  i32x8 g4 = {};
  __builtin_amdgcn_tensor_load_to_lds(g0, g1, g2, g3, g4, 0);
#else
  __builtin_amdgcn_tensor_load_to_lds(g0, g1, g2, g3, 0);
#endif
}
#endif  // HAVE_TDM

// =====================================================================
// fp32 WMMA GEMM:  C[M,N] = A[M,K] @ B   (BT=false: B is [K,N] row-major;
//                                         BT=true : B is [N,K] row-major)
// 256 threads = 8 waves; block tile 128x128, K-slab 32.
// LDS tiles are row-major with contiguous k and row stride LDST=36 dwords
// (144 B: 16-B aligned for float4 staging, bank-decorrelated for b64
// fragment reads). Each WMMA fragment is ONE aligned ds_load_b64.
// =====================================================================
template <bool BT>
__global__ __launch_bounds__(256, 2)
void gemm_wmma_f32(const float* __restrict__ A, const float* __restrict__ B,
                   float* __restrict__ C, int M, int N, int K) {
  constexpr int BM = 128, BN = 128, BK = 32, LDST = 36;
  __shared__ float Alds[BM * LDST];   // Alds[m*LDST + k]
  __shared__ float Blds[BN * LDST];   // Blds[n*LDST + k]

  const int tid  = threadIdx.x;
  const int lane = tid & 31;
  const int wave = tid >> 5;
  const int wm   = (wave & 3) * 32;   // wave M offset inside block tile
  const int wn   = (wave >> 2) * 64;  // wave N offset inside block tile
  const int bm   = blockIdx.x * BM;   // M fast-varying -> lm_w streamed ~once,
  const int bn   = blockIdx.y * BN;   // mixed (16MB) stays resident in 192MB L2

  const int am   = lane & 15;         // fragment row-in-subtile / col-in-subtile
  const int koff = (lane >> 4) << 1;  // half-wave covers K+{0,1} vs K+{2,3}

  // hoisted LDS fragment row bases
  const float* arow0 = Alds + (wm + am) * LDST;
  const float* arow1 = Alds + (wm + 16 + am) * LDST;
  const float* brow0 = Blds + (wn + am) * LDST;
  const float* brow1 = Blds + (wn + 16 + am) * LDST;
  const float* brow2 = Blds + (wn + 32 + am) * LDST;
  const float* brow3 = Blds + (wn + 48 + am) * LDST;

  v8f acc[2][4] = {};

  for (int k0 = 0; k0 < K; k0 += BK) {
#if HAVE_TDM
    // ---- DMA staging via Tensor Data Mover (wave 0 issues; EXEC ignored) ----
    if (wave == 0) {
      tdm_load_tile_2d((uint32_t)(uintptr_t)&Alds[0],
                       A + (size_t)bm * K + k0,
                       (uint32_t)(K - k0), (uint32_t)(M - bm), (uint32_t)K);
      if (BT)
        tdm_load_tile_2d((uint32_t)(uintptr_t)&Blds[0],
                         B + (size_t)bn * K + k0,
                         (uint32_t)(K - k0), (uint32_t)(N - bn), (uint32_t)K);
    }
    if (!BT) {
      // W is [K,N]: manual transposing stage into Blds[n][k]
#pragma unroll
      for (int p = 0; p < 4; ++p) {
        int kk  = (tid >> 5) + p * 8;      // 0..31
        int c4b = (tid & 31) * 4;          // 0..124
        int gn  = bn + c4b;
        const float* src = B + (size_t)(k0 + kk) * N + gn;
        float4 v = {0.f, 0.f, 0.f, 0.f};
        if (gn + 3 < N) v = *(const float4*)src;
        else {
          if (gn + 0 < N) v.x = src[0];
          if (gn + 1 < N) v.y = src[1];
          if (gn + 2 < N) v.z = src[2];
          if (gn + 3 < N) v.w = src[3];
        }
        Blds[(c4b + 0) * LDST + kk] = v.x;
        Blds[(c4b + 1) * LDST + kk] = v.y;
        Blds[(c4b + 2) * LDST + kk] = v.z;
        Blds[(c4b + 3) * LDST + kk] = v.w;
      }
    }
    if (wave == 0) __builtin_amdgcn_s_wait_tensorcnt(0);
    __syncthreads();
#else
    // ---- manual staging fallback (coalesced float4 loads) ----
#pragma unroll
    for (int p = 0; p < 4; ++p) {
      int r  = (tid >> 3) + p * 32;        // 0..127
      int c4 = (tid & 7) * 4;              // 0..28
      int gm = bm + r;
      float4 v = {0.f, 0.f, 0.f, 0.f};
      if (gm < M) v = *(const float4*)(A + (size_t)gm * K + k0 + c4);
      *(float4*)&Alds[r * LDST + c4] = v;  // 144B row stride -> 16B aligned
    }
    if (BT) {
#pragma unroll
      for (int p = 0; p < 4; ++p) {
        int r  = (tid >> 3) + p * 32;
        int c4 = (tid & 7) * 4;
        int gn = bn + r;
        float4 v = {0.f, 0.f, 0.f, 0.f};
        if (gn < N) v = *(const float4*)(B + (size_t)gn * K + k0 + c4);
        *(float4*)&Blds[r * LDST + c4] = v;
      }
    } else {
#pragma unroll
      for (int p = 0; p < 4; ++p) {
        int kk  = (tid >> 5) + p * 8;
        int c4b = (tid & 31) * 4;
        int gn  = bn + c4b;
        const float* src = B + (size_t)(k0 + kk) * N + gn;
        float4 v = {0.f, 0.f, 0.f, 0.f};
        if (gn + 3 < N) v = *(const float4*)src;
        else {
          if (gn + 0 < N) v.x = src[0];
          if (gn + 1 < N) v.y = src[1];
          if (gn + 2 < N) v.z = src[2];
          if (gn + 3 < N) v.w = src[3];
        }
        Blds[(c4b + 0) * LDST + kk] = v.x;
        Blds[(c4b + 1) * LDST + kk] = v.y;
        Blds[(c4b + 2) * LDST + kk] = v.z;
        Blds[(c4b + 3) * LDST + kk] = v.w;
      }
    }
    if (k0 + BK < K) {                      // global_prefetch_b8 next K-slab
      int gm = bm + (tid >> 3);
      if (gm < M) __builtin_prefetch(A + (size_t)gm * K + k0 + BK, 0, 1);
    }
    __syncthreads();
#endif

    // ---- 8 WMMAs per K-step of 4; fragments = single ds_load_b64 each ----
#pragma unroll
    for (int kk = 0; kk < BK; kk += 4) {
      const int ak = kk + koff;
      v2f afrag[2], bfrag[4];
      afrag[0] = *(const v2f*)(arow0 + ak);
      afrag[1] = *(const v2f*)(arow1 + ak);
      bfrag[0] = *(const v2f*)(brow0 + ak);
      bfrag[1] = *(const v2f*)(brow1 + ak);
      bfrag[2] = *(const v2f*)(brow2 + ak);
      bfrag[3] = *(const v2f*)(brow3 + ak);
#pragma unroll
      for (int i = 0; i < 2; ++i)
#pragma unroll
        for (int j = 0; j < 4; ++j)
          acc[i][j] = wmma_f32_16x16x4(afrag[i], bfrag[j], acc[i][j]);
    }
    __syncthreads();
  }

  // ---- store: C/D layout — VGPR v: lanes 0-15 -> M=v, lanes 16-31 -> M=v+8 ----
#pragma unroll
  for (int i = 0; i < 2; ++i) {
    const int msub = bm + wm + i * 16 + ((lane >> 4) << 3);
#pragma unroll
    for (int j = 0; j < 4; ++j) {
      const int n = bn + wn + j * 16 + am;
      if (n < N) {
#pragma unroll
        for (int v = 0; v < 8; ++v) {
          const int m = msub + v;
          if (m < M) C[(size_t)m * N + n] = acc[i][j][v];
        }
      }
    }
  }
}

// =====================================================================
// Segment means: seg_mean[t][b][d] = mean_l hidden[b][t*64+l][d]
// =====================================================================
__global__ void seg_mean_kernel(const float* __restrict__ hidden,
                                float* __restrict__ seg_mean,
                                int Bsz, int S, int D) {
  const int t = blockIdx.x, b = blockIdx.y;
  for (int d = threadIdx.x; d < D; d += blockDim.x) {
    const float* base = hidden + ((size_t)b * S + (size_t)t * SEGSZ) * D + d;
    float s = 0.f;
#pragma unroll 4
    for (int l = 0; l < SEGSZ; ++l) s += base[(size_t)l * D];
    seg_mean[((size_t)t * Bsz + b) * D + d] = s * (1.0f / SEGSZ);
  }
}

// score[t] = mean_b( seg_mean[t][b] . sel_w ) + sel_b
__global__ void seg_score_kernel(const float* __restrict__ seg_mean,
                                 const float* __restrict__ sel_w,
                                 const float* __restrict__ sel_b,
                                 float* __restrict__ score, int Bsz, int D) {
  const int t = blockIdx.x;
  __shared__ float red[256];
  float s = 0.f;
  for (int i = threadIdx.x; i < Bsz * D; i += blockDim.x)
    s += seg_mean[(size_t)t * Bsz * D + i] * sel_w[i % D];
  red[threadIdx.x] = s;
  __syncthreads();
  for (int o = 128; o > 0; o >>= 1) {
    if ((int)threadIdx.x < o) red[threadIdx.x] += red[threadIdx.x + o];
    __syncthreads();
  }
  if (threadIdx.x == 0) score[t] = red[0] / (float)Bsz + sel_b[0];
}

// =====================================================================
// Serial append/prune state machine (32 tiny steps; one lane).
// Records, per step t, the buffer BEFORE appending segment t:
//   rec_cnt[t] = #valid slots (valid slots are always a prefix)
//   rec_segid[t*CAPN+n] = source segment id of slot n (-1 if invalid)
// =====================================================================
__global__ void state_machine_kernel(const float* __restrict__ score,
                                     int* __restrict__ rec_segid,
                                     int* __restrict__ rec_cnt, int n_seg) {
  if (threadIdx.x != 0 || blockIdx.x != 0) return;
  float sc[CAPN]; int sid[CAPN]; bool valid[CAPN];
  for (int n = 0; n < CAPN; ++n) { sc[n] = 0.f; sid[n] = -1; valid[n] = false; }
  int cnt = 0;
  for (int t = 0; t < n_seg; ++t) {
    rec_cnt[t] = cnt;
    for (int n = 0; n < CAPN; ++n) rec_segid[t * CAPN + n] = valid[n] ? sid[n] : -1;
    // append
    sc[cnt] = score[t]; sid[cnt] = t; valid[cnt] = true; ++cnt;
    if (cnt > MAXCACHED) {
      bool any = false; bool elig[CAPN];
      for (int n = 0; n < CAPN; ++n) { elig[n] = valid[n] && (sc[n] >= THRESH); any |= elig[n]; }
      int am = 0; float best = -3.0e38f;
      for (int n = 0; n < CAPN; ++n) {
        float m = valid[n] ? sc[n] : NEGV;
        if (m > best) { best = m; am = n; }          // first max (jnp.argmax)
      }
      float eff[CAPN];
      for (int n = 0; n < CAPN; ++n) {
        bool keep = any ? elig[n] : (n == am);
        eff[n] = keep ? sc[n] : NEGV;
      }
      // top_k(TOPK) descending, lower index wins ties (selection by first max)
      float nsc[CAPN]; int nid[CAPN]; bool nval[CAPN]; bool used[CAPN];
      for (int n = 0; n < CAPN; ++n) { nsc[n] = 0.f; nid[n] = -1; nval[n] = false; used[n] = false; }
      int ncnt = 0;
      for (int r = 0; r < TOPK; ++r) {
        int bi = 0; float bv = -3.0e38f;
        for (int n = 0; n < CAPN; ++n)
          if (!used[n] && eff[n] > bv) { bv = eff[n]; bi = n; }
        used[bi] = true;
        if (bv > (NEGV * 0.5f)) { nsc[r] = bv; nid[r] = sid[bi]; nval[r] = true; ++ncnt; }
      }
      for (int n = 0; n < CAPN; ++n) { sc[n] = nsc[n]; sid[n] = nid[n]; valid[n] = nval[n]; }
      cnt = ncnt;  // kept entries form a prefix (eff sorted descending)
    }
  }
}

// =====================================================================
// Mixing: one wave32 per position p = b*S + s.
//   att_n = scale * <q[p], seg_mean[sid_n][b]>; masked softmax over valid
//   mixed = cnt>=1 ? gate*chunk + (1-gate)*wh : chunk
// =====================================================================
__global__ __launch_bounds__(256)
void mix_kernel(const float* __restrict__ hidden, const float* __restrict__ q,
                const float* __restrict__ seg_mean,
                const int* __restrict__ rec_segid, const int* __restrict__ rec_cnt,
                const float* __restrict__ online_bias,
                float* __restrict__ mixed, int Bsz, int S, int D) {
  const int gw   = (int)((blockIdx.x * blockDim.x + threadIdx.x) >> 5);
  const int lane = threadIdx.x & 31;
  if (gw >= Bsz * S) return;
  const int b = gw / S, s = gw - b * S, t = s / SEGSZ;

  const float gate  = 1.0f / (1.0f + __expf(-online_bias[0]));
  const float scale = rsqrtf((float)D);
  const float* qrow = q + (size_t)gw * D;
  const float* hrow = hidden + (size_t)gw * D;
  const int cnt = rec_cnt[t];

  int   sid[CAPN];
  float att[CAPN];
#pragma unroll
  for (int n = 0; n < CAPN; ++n) {
    int g = rec_segid[t * CAPN + n];
    sid[n] = (g < 0) ? 0 : g;
    float part = 0.f;
    if (n < cnt) {
      const float* mrow = seg_mean + ((size_t)sid[n] * Bsz + b) * D;
      for (int d = lane; d < D; d += 32) part += qrow[d] * mrow[d];
    }
    att[n] = part;
  }
  // wave32 butterfly reduce all slot partials
#pragma unroll
  for (int off = 16; off > 0; off >>= 1)
#pragma unroll
    for (int n = 0; n < CAPN; ++n) att[n] += __shfl_xor(att[n], off, 32);

  float mx = -3.0e38f;
#pragma unroll
  for (int n = 0; n < CAPN; ++n) {
    att[n] *= scale;
    if (n < cnt) mx = fmaxf(mx, att[n]);
  }
  float denom = 0.f;
#pragma unroll
  for (int n = 0; n < CAPN; ++n) {
    float e = (n < cnt) ? __expf(att[n] - mx) : 0.f;  // invalid slots ~= exp(NEG)
    att[n] = e; denom += e;
  }
  const float inv    = (cnt > 0) ? (1.0f / denom) : 0.f;
  const bool use_mix = (cnt >= 1);  // MIN_HIST

  for (int d = lane; d < D; d += 32) {
    float wh = 0.f;
#pragma unroll
    for (int n = 0; n < CAPN; ++n)
      if (n < cnt) wh += att[n] * seg_mean[((size_t)sid[n] * Bsz + b) * D + d];
    wh *= inv;
    const float c = hrow[d];
    mixed[(size_t)gw * D + d] = use_mix ? (gate * c + (1.0f - gate) * wh) : c;
  }
}

// =====================================================================
extern "C" void kernel_launch(void* const* d_in, const int* in_sizes, int n_in,
                              void* d_out, int out_size, void* d_ws, size_t ws_size,
                              hipStream_t stream) {
  const float* hidden      = (const float*)d_in[0];
  const float* W           = (const float*)d_in[1];
  const float* online_bias = (const float*)d_in[2];
  const float* sel_w       = (const float*)d_in[3];
  const float* sel_b       = (const float*)d_in[4];
  const float* lm_w        = (const float*)d_in[5];

  const int D     = in_sizes[3];            // 1024
  const int V     = in_sizes[5] / D;        // 50257
  const int Bsz   = 2;                      // per reference setup
  const int MS    = in_sizes[0] / D;        // B*S = 4096
  const int S     = MS / Bsz;               // 2048
  const int n_seg = S / SEGSZ;              // 32

  // workspace carve-up (~33 MB)
  float* q         = (float*)d_ws;
  float* mixed     = q + (size_t)MS * D;
  float* seg_mean  = mixed + (size_t)MS * D;
  float* seg_score = seg_mean + (size_t)n_seg * Bsz * D;
  int*   rec_segid = (int*)(seg_score + n_seg);
  int*   rec_cnt   = rec_segid + n_seg * CAPN;

  // 1) q = hidden @ W   (hoisted out of the scan; W is position-invariant)
  gemm_wmma_f32<false><<<dim3(MS / 128, (D + 127) / 128), 256, 0, stream>>>(
      hidden, W, q, MS, D, D);
  // 2) per-segment means (parallel — buffer holds raw-chunk means)
  seg_mean_kernel<<<dim3(n_seg, Bsz), 256, 0, stream>>>(hidden, seg_mean, Bsz, S, D);
  // 3) per-segment selection scores
  seg_score_kernel<<<n_seg, 256, 0, stream>>>(seg_mean, sel_w, sel_b, seg_score, Bsz, D);
  // 4) tiny serial append/prune/top-k state machine -> per-step buffer records
  state_machine_kernel<<<1, 32, 0, stream>>>(seg_score, rec_segid, rec_cnt, n_seg);
  // 5) attention mixing over cached means (one wave32 per position)
  mix_kernel<<<(MS * 32 + 255) / 256, 256, 0, stream>>>(
      hidden, q, seg_mean, rec_segid, rec_cnt, online_bias, mixed, Bsz, S, D);
  // 6) logits = mixed @ lm_w^T  — HBM-store-bound; M fast-varying keeps mixed
  //    resident in L2 while lm_w streams from HBM ~once.
  gemm_wmma_f32<true><<<dim3(MS / 128, (V + 127) / 128), 256, 0, stream>>>(
      mixed, lm_w, (float*)d_out, MS, V, D);
}